// FuzzyLayer_13091060318828
// MI455X (gfx1250) — compile-verified
//
#include <hip/hip_runtime.h>

typedef __attribute__((ext_vector_type(2))) float v2f;
typedef __attribute__((ext_vector_type(8))) float v8f;

#define KTOT 1024              // I*O = 32*32 parameter terms per sample
#define SAMPLES_PER_BLOCK 128  // 8 waves * 16 samples/wave

__global__ __launch_bounds__(256)
void fuzzy_wmma_kernel(const float* __restrict__ x,
                       const float* __restrict__ fd,
                       const float* __restrict__ sigma,
                       float* __restrict__ out, int n) {
    // LDS table: tab4[i] = { a_{2i}, b_{2i}, a_{2i+1}, b_{2i+1} }
    // where a_k = 1/sigma_k^2, b_k = -fd_k * a_k, so term = sqrt(fma(x, a, b))
    __shared__ float4 tab4[KTOT / 2];

    for (int i = threadIdx.x; i < KTOT / 2; i += 256) {
        int k0 = 2 * i;
        float s0 = sigma[k0],     f0 = fd[k0];
        float s1 = sigma[k0 + 1], f1 = fd[k0 + 1];
        float a0 = 1.0f / (s0 * s0);
        float a1 = 1.0f / (s1 * s1);
        tab4[i] = make_float4(a0, -f0 * a0, a1, -f1 * a1);
    }
    __syncthreads();

    const int lane = threadIdx.x & 31;
    const int wave = threadIdx.x >> 5;
    const int half = lane >> 4;   // 0: K+{0,1}, 1: K+{2,3}   (f32 A-matrix layout)
    const int col  = lane & 15;   // sample row M within the 16-sample tile
    const int base = blockIdx.x * SAMPLES_PER_BLOCK + wave * 16;

    int xi = base + col;
    float xv = x[xi < n ? xi : 0];

    v2f bones; bones.x = 1.0f; bones.y = 1.0f;   // all-ones B => D[m,n] = sum_k A[m,k] + C
    v8f acc0 = {};
    v8f acc1 = {};

    // 256 WMMA steps of K=4 cover all 1024 terms; two accumulators break the
    // D->C dependence chain so sqrt/FMA work pipelines under the matrix op.
    #pragma unroll 4
    for (int t = 0; t < 256; t += 2) {
        float4 ab0 = tab4[((t    ) << 1) | half];  // ds_load_b128, 16B aligned
        float4 ab1 = tab4[((t + 1) << 1) | half];
        v2f a, b;
        a.x = __builtin_amdgcn_sqrtf(__builtin_fmaf(xv, ab0.x, ab0.y)); // k = 4t+2h+0
        a.y = __builtin_amdgcn_sqrtf(__builtin_fmaf(xv, ab0.z, ab0.w)); // k = 4t+2h+1
        b.x = __builtin_amdgcn_sqrtf(__builtin_fmaf(xv, ab1.x, ab1.y));
        b.y = __builtin_amdgcn_sqrtf(__builtin_fmaf(xv, ab1.z, ab1.w));
        acc0 = __builtin_amdgcn_wmma_f32_16x16x4_f32(false, a, false, bones,
                                                     (short)0, acc0, false, false);
        acc1 = __builtin_amdgcn_wmma_f32_16x16x4_f32(false, b, false, bones,
                                                     (short)0, acc1, false, false);
    }

    // D layout: VGPR r = row r (lanes 0-15) / row r+8 (lanes 16-31), all 16
    // columns identical (B was all ones). Lane with col==r owns sample r+8*half.
    if (col < 8) {
        float s = 0.0f;
        #pragma unroll
        for (int r = 0; r < 8; ++r)
            if (col == r) s = acc0[r] + acc1[r];
        int oi = base + col + (half << 3);
        if (oi < n) {
            float e = __expf(-s);          // exp(-NaN) = NaN propagates
            out[oi] = (e != e) ? x[oi] : e; // reference NaN fallback
        }
    }
}

extern "C" void kernel_launch(void* const* d_in, const int* in_sizes, int n_in,
                              void* d_out, int out_size, void* d_ws, size_t ws_size,
                              hipStream_t stream) {
    const float* x  = (const float*)d_in[0];
    const float* fd = (const float*)d_in[1];
    const float* sg = (const float*)d_in[2];
    float* out = (float*)d_out;
    int n = in_sizes[0];
    int grid = (n + SAMPLES_PER_BLOCK - 1) / SAMPLES_PER_BLOCK;
    fuzzy_wmma_kernel<<<grid, 256, 0, stream>>>(x, fd, sg, out, n);
}